// DigitCaps_10144712753340
// MI455X (gfx1250) — compile-verified
//
#include <hip/hip_runtime.h>
#include <hip/hip_bf16.h>

typedef float v2f __attribute__((ext_vector_type(2)));
typedef float v4f __attribute__((ext_vector_type(4)));
typedef float v8f __attribute__((ext_vector_type(8)));

constexpr int kB    = 256;   // batch
constexpr int kI    = 2048;  // input capsules
constexpr int kC    = 10;    // output capsules
constexpr int kDin  = 8;
constexpr int kDout = 16;
constexpr int kCO   = kC * kDout;        // 160 floats per (b)
constexpr int kTileB   = 16;             // batches per block (WMMA N)
constexpr int kNChunk  = 16;             // i-chunks (partial reduction blocks)
constexpr int kWaves   = 4;              // waves per route block
constexpr int kIChunk  = kI / kNChunk;   // 128
constexpr int kIPerWave = kIChunk / kWaves; // 32

// ---------------------------------------------------------------------------
// zero Vsum (40960 floats); grid 160 x 256 covers it exactly
__global__ __launch_bounds__(256) void caps_zero(float* __restrict__ p) {
  p[(size_t)blockIdx.x * 256 + threadIdx.x] = 0.0f;
}

// ---------------------------------------------------------------------------
// One routing pass: for this (b-tile, i-chunk) block, compute
//   logits  b_ij[b,i,c] = u_hat[b,i,c,:] . Vsum[b,c,:]   (Vsum=0 -> uniform)
//   c_ij    = softmax_c(b_ij)
//   partial s_j[b,c,:] += sum_{i in chunk} c_ij * u_hat[b,i,c,:]
// u_hat tiles (16 b x 16 o) are produced by two chained
// v_wmma_f32_16x16x4_f32 (K=4 each, covering d=0..7):
//   A[m=o, k=d] = W[i,c,o,d]   (16x4)
//   B[k=d, n=b] = u[b0+n,i,d]  (4x16)
// D layout: lane = b (halves mirror), VGPR r = o (0..7 / 8..15 by lane half).
__global__ __launch_bounds__(128) void caps_route(
    const float* __restrict__ u, const float* __restrict__ W,
    const float* __restrict__ vsum, float* __restrict__ s_partial)
{
  __shared__ float vs_lds[kTileB * kCO];          // 10 KB  Vsum tile
  __shared__ float red[kWaves][kTileB * kCO];     // 40 KB  cross-wave reduce

  const int bt    = blockIdx.x;       // 0..15  (batch tile)
  const int chunk = blockIdx.y;       // 0..15  (i chunk)
  const int tid   = threadIdx.x;
  const int wave  = tid >> 5;
  const int lane  = tid & 31;
  const int half  = lane >> 4;        // K-half / o-half selector
  const int nl    = lane & 15;        // M index for A (=o), N index for B/D (=b)
  const int b0    = bt * kTileB;
  const int d0    = 2 * half;         // this lane's d-pair within a K=4 step

  // stage Vsum[b0..b0+15, :, :] into LDS (contiguous in global)
  for (int e = tid; e < kTileB * kCO; e += 128)
    vs_lds[e] = vsum[(size_t)b0 * kCO + e];
  __syncthreads();

  const v8f vzero = {0.f, 0.f, 0.f, 0.f, 0.f, 0.f, 0.f, 0.f};
  v8f acc[kC];
  #pragma unroll
  for (int c = 0; c < kC; ++c) acc[c] = vzero;

  const int i_base = chunk * kIChunk + wave * kIPerWave;

  for (int ii = 0; ii < kIPerWave; ++ii) {
    const int i = i_base + ii;

    // B operand: u[b0+nl, i, d0..d0+1] and [d0+4..d0+5]
    const float* up = u + ((size_t)(b0 + nl) * kI + i) * kDin + d0;
    const v2f bu_lo = *(const v2f*)(up);
    const v2f bu_hi = *(const v2f*)(up + 4);

    v8f   tile[kC];
    float logit[kC];
    #pragma unroll
    for (int c = 0; c < kC; ++c) {
      // A operand: W[i, c, nl, d0..d0+1] and [d0+4..d0+5]
      const float* wp = W + (((size_t)i * kC + c) * kDout + nl) * kDin + d0;
      const v2f aw_lo = *(const v2f*)(wp);
      const v2f aw_hi = *(const v2f*)(wp + 4);

      v8f t = __builtin_amdgcn_wmma_f32_16x16x4_f32(
                  false, aw_lo, false, bu_lo, (short)0, vzero, false, false);
      t     = __builtin_amdgcn_wmma_f32_16x16x4_f32(
                  false, aw_hi, false, bu_hi, (short)0, t,     false, false);
      tile[c] = t;

      // agreement dot: this lane holds u_hat[b=nl, i, c, o=r+8*half]
      const float* vp = &vs_lds[(nl * kC + c) * kDout + 8 * half];
      float p = 0.f;
      #pragma unroll
      for (int r = 0; r < 8; ++r) p = fmaf(t[r], vp[r], p);
      p += __shfl_xor(p, 16, 32);   // combine o=0..7 with o=8..15 (same b)
      logit[c] = p;
    }

    // softmax over the 10 output capsules (in-lane; halves duplicate)
    float m = logit[0];
    #pragma unroll
    for (int c = 1; c < kC; ++c) m = fmaxf(m, logit[c]);
    float esum = 0.f, ex[kC];
    #pragma unroll
    for (int c = 0; c < kC; ++c) { ex[c] = __expf(logit[c] - m); esum += ex[c]; }
    const float inv = 1.0f / esum;

    #pragma unroll
    for (int c = 0; c < kC; ++c) {
      const float wgt = ex[c] * inv;
      #pragma unroll
      for (int r = 0; r < 8; ++r) acc[c][r] = fmaf(wgt, tile[c][r], acc[c][r]);
    }
  }

  // park per-wave accumulators in LDS: red[wave][b_local][c][o]
  #pragma unroll
  for (int c = 0; c < kC; ++c) {
    float* dst = &red[wave][(nl * kC + c) * kDout + 8 * half];
    v4f lo = { acc[c][0], acc[c][1], acc[c][2], acc[c][3] };
    v4f hi = { acc[c][4], acc[c][5], acc[c][6], acc[c][7] };
    *(v4f*)(dst)     = lo;
    *(v4f*)(dst + 4) = hi;
  }
  __syncthreads();

  // cross-wave sum and deterministic partial write: s_partial[chunk][b][c][o]
  float* outp = s_partial + ((size_t)chunk * kB + b0) * kCO;
  for (int e4 = tid; e4 < (kTileB * kCO) / 4; e4 += 128) {
    const int e = e4 * 4;
    v4f r = *(v4f*)&red[0][e];
    r += *(v4f*)&red[1][e];
    r += *(v4f*)&red[2][e];
    r += *(v4f*)&red[3][e];
    *(v4f*)&outp[e] = r;
  }
}

// ---------------------------------------------------------------------------
// Reduce the 16 chunk partials, squash, then either fold v into Vsum
// (iterations 0,1) or emit the final output (iteration 2).
// 16 lanes per (b,c) pair; o = lane&15.
__global__ __launch_bounds__(128) void caps_squash(
    const float* __restrict__ s_partial, float* __restrict__ vsum,
    float* __restrict__ out, const int is_final)
{
  const int tid  = threadIdx.x;
  const int o    = tid & 15;
  const int pair = blockIdx.x * 8 + (tid >> 4);   // 0..2559
  const int b    = pair / kC;
  const int c    = pair % kC;

  float s = 0.f;
  #pragma unroll
  for (int ch = 0; ch < kNChunk; ++ch)
    s += s_partial[((size_t)ch * kB + b) * kCO + c * kDout + o];

  float sq = s * s;                       // ||s||^2 over the 16 o-lanes
  #pragma unroll
  for (int mask = 8; mask >= 1; mask >>= 1) sq += __shfl_xor(sq, mask, 32);

  const float v = s * (sq / ((1.0f + sq) * sqrtf(sq + 1e-9f)));
  const size_t idx = ((size_t)b * kC + c) * kDout + o;
  if (is_final) out[idx] = v;
  else          vsum[idx] += v;           // one writer per element
}

// ---------------------------------------------------------------------------
extern "C" void kernel_launch(void* const* d_in, const int* in_sizes, int n_in,
                              void* d_out, int out_size, void* d_ws, size_t ws_size,
                              hipStream_t stream) {
  const float* u = (const float*)d_in[0];   // [256, 2048, 8]
  const float* W = (const float*)d_in[1];   // [2048, 10, 16, 8]
  float* out = (float*)d_out;               // [256, 10, 16]

  float* vsum      = (float*)d_ws;                    // 40960 floats
  float* s_partial = vsum + (size_t)kB * kCO;         // 16*256*160 floats (2.6 MB)

  caps_zero<<<(kB * kCO) / 256, 256, 0, stream>>>(vsum);

  for (int it = 0; it < 3; ++it) {
    caps_route<<<dim3(kB / kTileB, kNChunk), 128, 0, stream>>>(u, W, vsum, s_partial);
    caps_squash<<<(kB * kC) / 8, 128, 0, stream>>>(s_partial, vsum, out,
                                                   (it == 2) ? 1 : 0);
  }
}